// crossCorrelation2D_38654705664136
// MI455X (gfx1250) — compile-verified
//
#include <hip/hip_runtime.h>

// CDNA5 / gfx1250, wave32. Fused 9x9 local cross-correlation loss.
// Vertical box sum implemented as band-matrix multiply on V_WMMA_F32_16X16X4_F32.

typedef __attribute__((ext_vector_type(2))) float v2f;
typedef __attribute__((ext_vector_type(8))) float v8f;

constexpr int IMG_W  = 512;
constexpr int IMG_H  = 512;
constexpr int BATCH  = 32;
constexpr int TILE   = 32;          // output tile (both dims)
constexpr int RGN    = TILE + 8;    // input region incl. 4-halo each side
constexpr int HS     = TILE + 1;    // padded LDS row stride (bank-conflict free)
constexpr int NBLK   = (IMG_W / TILE) * (IMG_H / TILE) * BATCH;  // 8192
constexpr float KSZ  = 81.0f;

__global__ __launch_bounds__(256)
void cc_tile_kernel(const float* __restrict__ inp,
                    const float* __restrict__ tgt,
                    float* __restrict__ partial)
{
    // hsum: horizontal 9-tap sums of {T, I, TT, II, IT}, rows 0..39 x cols 0..31
    __shared__ float sh_hsum[5][RGN][HS];        // 5*40*33*4 = 26400 B
    // buf: phase1/2 raw i,T (2*1600 floats); phase3/4 vertical sums (5*32*33)
    __shared__ float sh_buf[5 * TILE * HS];      // 21120 B
    __shared__ float sh_red[256];                // 1024 B

    const int tid  = threadIdx.x;
    const int row0 = blockIdx.y * TILE - 4;
    const int col0 = blockIdx.x * TILE - 4;
    const size_t imgOff = (size_t)blockIdx.z * (IMG_H * IMG_W);
    const float* iB = inp + imgOff;
    const float* tB = tgt + imgOff;

    float* rawI = sh_buf;              // 1600 floats
    float* rawT = sh_buf + RGN * RGN;  // 1600 floats (3200 total < 5280)

    // ---- Phase 1: load 40x40 region, zero-padded (SAME conv), rescale target.
    for (int idx = tid; idx < RGN * RGN; idx += 256) {
        int r = idx / RGN, c = idx % RGN;
        int gr = row0 + r, gc = col0 + c;
        float iv = 0.0f, tv = 0.0f;
        if ((unsigned)gr < (unsigned)IMG_H && (unsigned)gc < (unsigned)IMG_W) {
            size_t o = (size_t)gr * IMG_W + gc;
            iv = iB[o];
            tv = tB[o];
        }
        rawI[idx] = iv;
        rawT[idx] = (tv + 1.0f) * 0.5f;   // rescale [-1,1] -> [0,1]
    }
    __syncthreads();

    // ---- Phase 2: horizontal 9-tap sums for all 5 quantities (40 rows x 32 cols).
    for (int idx = tid; idx < RGN * TILE; idx += 256) {
        int r = idx >> 5;       // 0..39
        int c = idx & 31;       // 0..31
        const float* ri = rawI + r * RGN + c;
        const float* rt = rawT + r * RGN + c;
        float sT = 0.f, sI = 0.f, sTT = 0.f, sII = 0.f, sIT = 0.f;
#pragma unroll
        for (int d = 0; d < 9; ++d) {
            float iv = ri[d], tv = rt[d];
            sT += tv; sI += iv;
            sTT += tv * tv; sII += iv * iv; sIT += iv * tv;
        }
        sh_hsum[0][r][c] = sT;
        sh_hsum[1][r][c] = sI;
        sh_hsum[2][r][c] = sTT;
        sh_hsum[3][r][c] = sII;
        sh_hsum[4][r][c] = sIT;
    }
    __syncthreads();  // hsum ready; raw region may now be overwritten

    // ---- Phase 3: vertical 9-tap sum as WMMA band-matmul.
    // O(16x16) = S(16x24, ones band: S[m][k]=1 iff m<=k<=m+8) * X(24x16 hsum strip),
    // accumulated over six K=4 chunks with V_WMMA_F32_16X16X4_F32.
    {
        const int wave = tid >> 5;
        const int lane = tid & 31;
        const int m  = lane & 15;            // A row / B,D column
        const int kb = (lane >> 4) << 1;     // K pair base: lanes<16 -> {0,1}, else {2,3}
        float* vsum = sh_buf;                // overlays raw region: 5 * 32 * 33
        for (int job = wave; job < 20; job += 8) {   // 5 quantities x 4 subtiles
            int q    = job >> 2;
            int sub  = job & 3;
            int subR = (sub >> 1) << 4;      // 0 or 16
            int subC = (sub & 1) << 4;       // 0 or 16
            v8f acc = {};
#pragma unroll
            for (int chunk = 0; chunk < 6; ++chunk) {
                int k0 = chunk * 4 + kb;
                v2f a, b;
                // A fragment (16x4 fp32): lane holds (m, k0) and (m, k0+1)
                a.x = (k0     >= m && k0     <= m + 8) ? 1.0f : 0.0f;
                a.y = (k0 + 1 >= m && k0 + 1 <= m + 8) ? 1.0f : 0.0f;
                // B fragment (4x16 fp32): lane holds (k0, n=m) and (k0+1, n=m)
                b.x = sh_hsum[q][subR + k0    ][subC + m];
                b.y = sh_hsum[q][subR + k0 + 1][subC + m];
                acc = __builtin_amdgcn_wmma_f32_16x16x4_f32(
                        /*neg_a=*/false, a, /*neg_b=*/false, b,
                        /*c_mod=*/(short)0, acc,
                        /*reuse_a=*/false, /*reuse_b=*/false);
            }
            // D layout: VGPR v holds row v (lanes 0-15) / row v+8 (lanes 16-31), col m
            int rb = (lane >> 4) << 3;       // 0 or 8
            float* vq = vsum + q * TILE * HS;
#pragma unroll
            for (int v = 0; v < 8; ++v)
                vq[(subR + rb + v) * HS + (subC + m)] = acc[v];
        }
    }
    __syncthreads();

    // ---- Phase 4: pointwise cc + deterministic block reduction.
    float local = 0.0f;
    const float* vsum = sh_buf;
    for (int idx = tid; idx < TILE * TILE; idx += 256) {
        int r = idx >> 5, c = idx & 31;
        int o = r * HS + c;
        float Ts  = vsum[0 * TILE * HS + o];
        float Is  = vsum[1 * TILE * HS + o];
        float TTs = vsum[2 * TILE * HS + o];
        float IIs = vsum[3 * TILE * HS + o];
        float ITs = vsum[4 * TILE * HS + o];
        float Ihat = Is * (1.0f / KSZ);
        float That = Ts * (1.0f / KSZ);
        float cross = ITs - Ihat * Ts - That * Is + That * Ihat * KSZ;
        float Tvar  = TTs - 2.0f * That * Ts + That * That * KSZ;
        float Ivar  = IIs - 2.0f * Ihat * Is + Ihat * Ihat * KSZ;
        local += cross * cross / (Tvar * Ivar + 1e-5f);
    }
    sh_red[tid] = local;
    __syncthreads();
    for (int s = 128; s > 0; s >>= 1) {
        if (tid < s) sh_red[tid] += sh_red[tid + s];
        __syncthreads();
    }
    if (tid == 0) {
        int blin = blockIdx.x + gridDim.x * (blockIdx.y + gridDim.y * blockIdx.z);
        partial[blin] = sh_red[0];
    }
}

// Single-block, fixed-order reduction of the 8192 partials -> scalar loss.
__global__ __launch_bounds__(256)
void cc_reduce_kernel(const float* __restrict__ partial, float* __restrict__ out)
{
    __shared__ float red[256];
    const int tid = threadIdx.x;
    constexpr int PER = NBLK / 256;   // 32
    float s = 0.0f;
    for (int i = 0; i < PER; ++i)
        s += partial[tid * PER + i];
    red[tid] = s;
    __syncthreads();
    for (int st = 128; st > 0; st >>= 1) {
        if (tid < st) red[tid] += red[tid + st];
        __syncthreads();
    }
    if (tid == 0)
        out[0] = -red[0] * (1.0f / (float)(BATCH * IMG_H * IMG_W));
}

extern "C" void kernel_launch(void* const* d_in, const int* in_sizes, int n_in,
                              void* d_out, int out_size, void* d_ws, size_t ws_size,
                              hipStream_t stream)
{
    (void)in_sizes; (void)n_in; (void)out_size; (void)ws_size;
    const float* inp = (const float*)d_in[0];   // (32,1,512,512) f32
    const float* tgt = (const float*)d_in[1];   // (32,1,512,512) f32
    // d_in[2] is the all-ones 9x9 filter; it is baked into the kernel math.
    float* out     = (float*)d_out;             // scalar f32
    float* partial = (float*)d_ws;              // NBLK * 4 B = 32 KB scratch

    dim3 grid(IMG_W / TILE, IMG_H / TILE, BATCH);  // 16 x 16 x 32
    cc_tile_kernel<<<grid, 256, 0, stream>>>(inp, tgt, partial);
    cc_reduce_kernel<<<1, 256, 0, stream>>>(partial, out);
}